// KuramotoOscillator_83829171683700
// MI455X (gfx1250) — compile-verified
//
#include <hip/hip_runtime.h>
#include <math.h>

typedef float v2f __attribute__((ext_vector_type(2)));
typedef float v8f __attribute__((ext_vector_type(8)));

#define TWO_PI_F 6.283185307179586476925f

// ---------------------------------------------------------------------------
// Generic fp32 GEMM via V_WMMA_F32_16X16X4_F32.
// C[M x N] = A[M x K] (row-major, stride lda) @ W[N x K]^T + bias, + activation.
// ACT: 0 = none, 1 = tanh, 2 = exact GELU.
// One wave computes one 16x16 output tile; 8 waves per block.
// A-frag layout (ISA 7.12.2, 32-bit A 16x4): lanes 0-15 hold K=k0,k0+1 for
// row M=lane; lanes 16-31 hold K=k0+2,k0+3. B-frag mirrored (4x16).
// ---------------------------------------------------------------------------
template <int ACT>
__global__ __launch_bounds__(256) void gemm_wmma(
    const float* __restrict__ A, int lda,
    const float* __restrict__ W,     // [N x K] row-major (torch Linear weight)
    const float* __restrict__ bias,  // [N]
    float* __restrict__ C, int ldc,
    int M, int N, int K) {
  const int lane = threadIdx.x & 31;
  const int wave = threadIdx.x >> 5;
  const int nTiles = N >> 4;
  const int totalTiles = (M >> 4) * nTiles;
  const int tile = blockIdx.x * 8 + wave;
  if (tile >= totalTiles) return;
  const int m0 = (tile / nTiles) << 4;
  const int n0 = (tile % nTiles) << 4;
  const int r = lane & 15;
  const int half = lane >> 4;

  const float* __restrict__ Arow = A + (size_t)(m0 + r) * lda;
  const float* __restrict__ Wrow = W + (size_t)(n0 + r) * K;

  v8f acc = {0.f, 0.f, 0.f, 0.f, 0.f, 0.f, 0.f, 0.f};
  for (int k0 = 0; k0 < K; k0 += 4) {
    const int ka = k0 + (half << 1);
    v2f a, b;
    a.x = Arow[ka];
    a.y = Arow[ka + 1];
    b.x = Wrow[ka];
    b.y = Wrow[ka + 1];
    acc = __builtin_amdgcn_wmma_f32_16x16x4_f32(
        /*neg_a=*/false, a, /*neg_b=*/false, b,
        /*c_mod=*/(short)0, acc, /*reuse_a=*/false, /*reuse_b=*/false);
  }

#pragma unroll
  for (int v = 0; v < 8; ++v) {
    const int row = m0 + (half << 3) + v;
    const int col = n0 + r;
    float val = acc[v] + bias[col];
    if (ACT == 1) val = tanhf(val);
    if (ACT == 2) val = 0.5f * val * (1.0f + erff(val * 0.70710678118654752f));
    C[(size_t)row * ldc + col] = val;
  }
}

// ---------------------------------------------------------------------------
// Per-row LayerNorm(D=128) + activation, in place. One 128-thread block / row.
// ---------------------------------------------------------------------------
template <int ACT>  // 1 = tanh, 2 = gelu
__global__ __launch_bounds__(128) void layernorm_act(
    float* __restrict__ X, const float* __restrict__ g,
    const float* __restrict__ b) {
  __shared__ float red[128];
  const int row = blockIdx.x;
  const int t = threadIdx.x;
  float v = X[(size_t)row * 128 + t];
  red[t] = v;
  __syncthreads();
  for (int s = 64; s > 0; s >>= 1) {
    if (t < s) red[t] += red[t + s];
    __syncthreads();
  }
  const float mean = red[0] * (1.0f / 128.0f);
  __syncthreads();
  const float d = v - mean;
  red[t] = d * d;
  __syncthreads();
  for (int s = 64; s > 0; s >>= 1) {
    if (t < s) red[t] += red[t + s];
    __syncthreads();
  }
  const float var = red[0] * (1.0f / 128.0f);
  float y = d * rsqrtf(var + 1e-5f) * g[t] + b[t];
  if (ACT == 1) y = tanhf(y);
  if (ACT == 2) y = 0.5f * y * (1.0f + erff(y * 0.70710678118654752f));
  X[(size_t)row * 128 + t] = y;
}

// ---------------------------------------------------------------------------
// Linear(64 -> 1) head. One wave32 per row; MODE 0 = identity, 1 = sigmoid.
// ---------------------------------------------------------------------------
template <int MODE>
__global__ __launch_bounds__(256) void head64(
    const float* __restrict__ X, const float* __restrict__ w,
    const float* __restrict__ b, float* __restrict__ out, int M) {
  const int lane = threadIdx.x & 31;
  const int row = blockIdx.x * 8 + (threadIdx.x >> 5);
  if (row >= M) return;
  const float* __restrict__ x = X + (size_t)row * 64;
  float s = x[lane] * w[lane] + x[lane + 32] * w[lane + 32];
  for (int o = 16; o > 0; o >>= 1) s += __shfl_xor(s, o, 32);
  if (lane == 0) {
    float v = s + b[0];
    if (MODE == 1) v = 1.0f / (1.0f + expf(-v));
    out[row] = v;
  }
}

// ---------------------------------------------------------------------------
// Phase init: atan2( sum_k x[k]*sin(k), sum_k x[k]*cos(k) ), D=128.
// ---------------------------------------------------------------------------
__global__ __launch_bounds__(256) void phase_init(
    const float* __restrict__ X, float* __restrict__ ph, int M) {
  const int lane = threadIdx.x & 31;
  const int row = blockIdx.x * 8 + (threadIdx.x >> 5);
  if (row >= M) return;
  const float* __restrict__ x = X + (size_t)row * 128;
  float ss = 0.f, cc = 0.f;
#pragma unroll
  for (int i = 0; i < 4; ++i) {
    const int k = lane + 32 * i;
    const float v = x[k];
    ss += v * sinf((float)k);
    cc += v * cosf((float)k);
  }
  for (int o = 16; o > 0; o >>= 1) {
    ss += __shfl_xor(ss, o, 32);
    cc += __shfl_xor(cc, o, 32);
  }
  if (lane == 0) ph[row] = atan2f(ss, cc);
}

// ---------------------------------------------------------------------------
// Per-step trig precompute (tiny; keeps cos/sin L2-resident for row kernels).
// ---------------------------------------------------------------------------
__global__ __launch_bounds__(256) void phase_trig(
    const float* __restrict__ ph, float* __restrict__ cp,
    float* __restrict__ sp, int n) {
  const int i = blockIdx.x * 256 + threadIdx.x;
  if (i < n) {
    const float p = ph[i];
    cp[i] = cosf(p);
    sp[i] = sinf(p);
  }
}

// ---------------------------------------------------------------------------
// One Kuramoto step for one row (b,i): both neighbor dots in a single pass
// over the adjacency row (A is L2-resident: 134 MB < 192 MB global L2).
// In-place phase update is safe: block gid touches only ph[gid].
// ---------------------------------------------------------------------------
__global__ __launch_bounds__(256) void kuramoto_step(
    const float* __restrict__ A, const float* __restrict__ cosP,
    const float* __restrict__ sinP, const float* __restrict__ natf,
    const float* __restrict__ lc, const float* __restrict__ gc,
    float* __restrict__ ph, int N) {
  const int gid = blockIdx.x;           // b*N + i
  const int b = gid >> 12;              // N == 4096
  const float* __restrict__ arow = A + (size_t)gid * N;
  const float* __restrict__ cp = cosP + (size_t)b * N;
  const float* __restrict__ sp = sinP + (size_t)b * N;

  float ac = 0.f, as = 0.f;
  for (int j = threadIdx.x * 4; j < N; j += 256 * 4) {
    __builtin_prefetch(arow + j + 1024, 0, 0);  // speculative, safely dropped
    const float4 av = *(const float4*)(arow + j);
    const float4 cv = *(const float4*)(cp + j);
    const float4 sv = *(const float4*)(sp + j);
    ac += av.x * cv.x + av.y * cv.y + av.z * cv.z + av.w * cv.w;
    as += av.x * sv.x + av.y * sv.y + av.z * sv.z + av.w * sv.w;
  }
  for (int o = 16; o > 0; o >>= 1) {
    ac += __shfl_xor(ac, o, 32);
    as += __shfl_xor(as, o, 32);
  }
  __shared__ float rc[8], rs[8];
  const int lane = threadIdx.x & 31, wave = threadIdx.x >> 5;
  if (lane == 0) {
    rc[wave] = ac;
    rs[wave] = as;
  }
  __syncthreads();
  if (threadIdx.x == 0) {
    float nc = 0.f, ns = 0.f;
#pragma unroll
    for (int w = 0; w < 8; ++w) {
      nc += rc[w];
      ns += rs[w];
    }
    const float p = ph[gid];
    const float ce = atan2f(ns, nc + 1e-8f) - p;
    const float g = fminf(fmaxf(gc[0], 0.1f), 2.0f);
    const float dphi = natf[gid] + lc[gid] * g * sinf(ce);
    float np = p + 0.01f * dphi;
    np = fmodf(np, TWO_PI_F);
    if (np < 0.f) np += TWO_PI_F;
    ph[gid] = np;
  }
}

// ---------------------------------------------------------------------------
// Build sync_input [row][132] = [node_features(128) | cos, sin, cos2, sin2];
// also emits the phases output.
// ---------------------------------------------------------------------------
__global__ __launch_bounds__(128) void build_sync_input(
    const float* __restrict__ X, const float* __restrict__ ph,
    float* __restrict__ SI, float* __restrict__ outPh) {
  const int row = blockIdx.x;
  const int t = threadIdx.x;
  float* __restrict__ o = SI + (size_t)row * 132;
  o[t] = X[(size_t)row * 128 + t];
  if (t == 0) {
    const float p = ph[row];
    o[128] = cosf(p);
    o[129] = sinf(p);
    o[130] = cosf(2.f * p);
    o[131] = sinf(2.f * p);
    outPh[row] = p;
  }
}

// ---------------------------------------------------------------------------
// Per-batch order parameter + phase coherence (unbiased std, ddof=1).
// ---------------------------------------------------------------------------
__global__ __launch_bounds__(256) void batch_stats(
    const float* __restrict__ ph, float* __restrict__ orderOut,
    float* __restrict__ cohOut, int N) {
  const int b = blockIdx.x;
  const float* __restrict__ p = ph + (size_t)b * N;
  float sc = 0.f, ss = 0.f, sp = 0.f, sq = 0.f;
  for (int i = threadIdx.x; i < N; i += 256) {
    const float v = p[i];
    sc += cosf(v);
    ss += sinf(v);
    sp += v;
    sq += v * v;
  }
  for (int o = 16; o > 0; o >>= 1) {
    sc += __shfl_xor(sc, o, 32);
    ss += __shfl_xor(ss, o, 32);
    sp += __shfl_xor(sp, o, 32);
    sq += __shfl_xor(sq, o, 32);
  }
  __shared__ float rc[8], rs[8], rp[8], rq[8];
  const int lane = threadIdx.x & 31, wave = threadIdx.x >> 5;
  if (lane == 0) {
    rc[wave] = sc; rs[wave] = ss; rp[wave] = sp; rq[wave] = sq;
  }
  __syncthreads();
  if (threadIdx.x == 0) {
    float tc = 0.f, ts = 0.f, tp = 0.f, tq = 0.f;
#pragma unroll
    for (int w = 0; w < 8; ++w) {
      tc += rc[w]; ts += rs[w]; tp += rp[w]; tq += rq[w];
    }
    const float mc = tc / (float)N, ms = ts / (float)N;
    orderOut[b] = sqrtf(mc * mc + ms * ms);
    const float var = (tq - tp * tp / (float)N) / (float)(N - 1);
    cohOut[b] = 1.0f / (1.0f + sqrtf(var));
  }
}

// ---------------------------------------------------------------------------
// Host-side orchestration.
// ---------------------------------------------------------------------------
extern "C" void kernel_launch(void* const* d_in, const int* in_sizes, int n_in,
                              void* d_out, int out_size, void* d_ws,
                              size_t ws_size, hipStream_t stream) {
  (void)in_sizes; (void)n_in; (void)out_size; (void)ws_size;
  const int B = 2, N = 4096, D = 128, M = B * N;  // M = 8192

  const float* X   = (const float*)d_in[0];   // [2,4096,128]
  const float* Adj = (const float*)d_in[1];   // [2,4096,4096]
  const float* fw1 = (const float*)d_in[2];
  const float* fb1 = (const float*)d_in[3];
  const float* fg  = (const float*)d_in[4];
  const float* fbe = (const float*)d_in[5];
  const float* fw2 = (const float*)d_in[6];
  const float* fb2 = (const float*)d_in[7];
  const float* fw3 = (const float*)d_in[8];
  const float* fb3 = (const float*)d_in[9];
  const float* cw1 = (const float*)d_in[10];
  const float* cb1 = (const float*)d_in[11];
  const float* cw2 = (const float*)d_in[12];
  const float* cb2 = (const float*)d_in[13];
  const float* sw1 = (const float*)d_in[14];
  const float* sb1 = (const float*)d_in[15];
  const float* sg  = (const float*)d_in[16];
  const float* sbe = (const float*)d_in[17];
  const float* sw2 = (const float*)d_in[18];
  const float* sb2 = (const float*)d_in[19];
  const float* sw3 = (const float*)d_in[20];
  const float* sb3 = (const float*)d_in[21];
  const float* gc  = (const float*)d_in[22];

  float* out = (float*)d_out;
  float* outDesync = out;                 // 8192
  float* outOrder  = out + 8192;          // 2
  float* outPhases = out + 8194;          // 8192
  float* outNatF   = out + 16386;         // 8192
  float* outLC     = out + 24578;         // 8192
  float* outSI     = out + 32770;         // 8192*132 = 1081344
  float* outCoh    = out + 1114114;       // 2

  float* ws = (float*)d_ws;
  float* h1 = ws;                  // 8192*128
  float* h2 = h1 + 1048576;        // 8192*64
  float* c1 = h2 + 524288;         // 8192*64
  float* s1 = c1 + 524288;         // 8192*128
  float* s2 = s1 + 1048576;        // 8192*64
  float* ph = s2 + 524288;         // 8192
  float* cp = ph + 8192;           // 8192
  float* sp = cp + 8192;           // 8192

  // --- frequency network ---
  gemm_wmma<0><<<512, 256, 0, stream>>>(X, D, fw1, fb1, h1, 128, M, 128, 128);
  layernorm_act<1><<<M, 128, 0, stream>>>(h1, fg, fbe);
  gemm_wmma<1><<<256, 256, 0, stream>>>(h1, 128, fw2, fb2, h2, 64, M, 64, 128);
  head64<0><<<M / 8, 256, 0, stream>>>(h2, fw3, fb3, outNatF, M);

  // --- coupling network ---
  gemm_wmma<2><<<256, 256, 0, stream>>>(X, D, cw1, cb1, c1, 64, M, 64, 128);
  head64<1><<<M / 8, 256, 0, stream>>>(c1, cw2, cb2, outLC, M);

  // --- phase init + 60-step Kuramoto scan (A stays L2-resident) ---
  phase_init<<<M / 8, 256, 0, stream>>>(X, ph, M);
  for (int step = 0; step < 60; ++step) {
    phase_trig<<<M / 256, 256, 0, stream>>>(ph, cp, sp, M);
    kuramoto_step<<<M, 256, 0, stream>>>(Adj, cp, sp, outNatF, outLC, gc, ph, N);
  }

  // --- sync analyzer ---
  build_sync_input<<<M, 128, 0, stream>>>(X, ph, outSI, outPhases);
  gemm_wmma<0><<<512, 256, 0, stream>>>(outSI, 132, sw1, sb1, s1, 128, M, 128, 132);
  layernorm_act<2><<<M, 128, 0, stream>>>(s1, sg, sbe);
  gemm_wmma<2><<<256, 256, 0, stream>>>(s1, 128, sw2, sb2, s2, 64, M, 64, 128);
  head64<1><<<M / 8, 256, 0, stream>>>(s2, sw3, sb3, outDesync, M);

  // --- order parameter / coherence ---
  batch_stats<<<B, 256, 0, stream>>>(ph, outOrder, outCoh, N);
}